// TensorProductConvLayer_81578608820634
// MI455X (gfx1250) — compile-verified
//
#include <hip/hip_runtime.h>

typedef __attribute__((ext_vector_type(2))) float    v2f;
typedef __attribute__((ext_vector_type(8))) float    v8f;
typedef __attribute__((ext_vector_type(4))) unsigned v4u;
typedef __attribute__((ext_vector_type(8))) unsigned v8u;
typedef __attribute__((ext_vector_type(4))) int      v4i;

#define NNODES 50000
#define NEDGES 800000
#define MUL0   64
#define MUL1   32
#define DIMF   160   // 64 + 3*32
#define EAF    32
#define NEFD   128
#define NPATH  4

#define WAVES  2
#define TPB    (WAVES * 32)

// padded LDS row stride for the gathered node tile: TDM pad inserts 1 DWORD
// after every 32 data DWORDs -> 160 + 5 = 165 DWORDs per row (odd stride ->
// conflict-free column reads across the 16 rows).
#define XSTRIDE 165
#define XTILE   (16 * XSTRIDE)

// path normalization constants
#define A000 0.08838834764831845f   // 1/sqrt(128)
#define A110 0.125f                 // 1/sqrt(64)
#define A011 0.08838834764831845f   // 1/sqrt(128)
#define A101 0.125f                 // 1/sqrt(64)
#define INV_SQRT3 0.5773502691896258f

__device__ __forceinline__ v8f wmma_f32_4(v2f a, v2f b, v8f c) {
    // D = A(16x4 f32) * B(4x16 f32) + C(16x16 f32)
    return __builtin_amdgcn_wmma_f32_16x16x4_f32(
        /*neg_a=*/false, a, /*neg_b=*/false, b,
        /*c_mod=*/(short)0, c, /*reuse_a=*/false, /*reuse_b=*/false);
}

// LDS offset (in floats) of element (row r, col c) of the TDM-padded tile
__device__ __forceinline__ int xoff(int r, int c) {
    return XSTRIDE * r + c + (c >> 5);
}

__global__ __launch_bounds__(TPB)
void tpconv_edge_kernel(const float* __restrict__ nat,   // node_attr  [N,160]
                        const float* __restrict__ eat,   // edge_attr  [E,32]
                        const float* __restrict__ esh,   // edge_sh    [E,4]
                        const float* __restrict__ W000,  // [64,64]
                        const float* __restrict__ W011,  // [64,32]
                        const float* __restrict__ W101,  // [32,32]
                        const float* __restrict__ W110,  // [32,64]
                        const float* __restrict__ fw1,   // [32,128]
                        const float* __restrict__ fb1,   // [128]
                        const float* __restrict__ fw2,   // [128,4]
                        const float* __restrict__ fb2,   // [4]
                        const int*   __restrict__ eidx,  // [2,E]
                        float* __restrict__ acc,         // [N,160] accumulator (pre-zeroed)
                        float* __restrict__ cnt)         // [N] counts (pre-zeroed)
{
    __shared__ float xtS[WAVES][XTILE];         // TDM-gathered node rows (padded)
    __shared__ float htS[WAVES][16][NEFD + 1];  // relu(h) tile, 16 edges x 128
    __shared__ float shS[WAVES][16][4];         // edge_sh per tile
    __shared__ float wS [WAVES][16][4];         // MLP path weights per tile
    __shared__ int   srcS[WAVES][16];

    const int wave = threadIdx.x >> 5;
    const int lane = threadIdx.x & 31;
    const int r    = lane & 15;       // matrix row (A) / col (B,C,D)
    const int half = lane >> 4;
    const int ko   = half << 1;       // K sub-offset within a 4-step
    const long e0  = ((long)blockIdx.x * WAVES + wave) << 4;  // first edge of tile
    if (e0 >= NEDGES) return;         // E % 32 == 0 -> uniform per block

    // ---- issue TDM gather of the 16 dst node rows into LDS (async) ---------
    {
        const int dval = eidx[NEDGES + e0 + r];   // lanes 16-31 duplicate 0-15
        const unsigned ldsbase =
            (unsigned)(unsigned long long)(const void*)&xtS[wave][0];
        const unsigned long long ga = (unsigned long long)nat;

        // D# group0: count=1, gather_mode=1, 32-bit indices, lds/global addr, type=2
        v4u g0; 
        g0.x = 0xC0000001u;
        g0.y = ldsbase;
        g0.z = (unsigned)ga;
        g0.w = ((unsigned)(ga >> 32) & 0x01FFFFFFu) | 0x80000000u;
        // D# group1: data_size=4B, pad_enable, pad_interval=32 DW, pad_amount=1 DW,
        //            tensor_dim0=160, tensor_dim1=N, tile_dim0=160, tile_dim1=8,
        //            tensor_dim0_stride=160
        v8u g1;
        g1.s0 = 0x01120000u;                        // pads + data_size=2(4B)
        g1.s1 = (unsigned)(DIMF) << 16;             // tensor_dim0[15:0]
        g1.s2 = ((unsigned)NNODES & 0xFFFFu) << 16; // dim0 hi=0 | tensor_dim1 lo
        g1.s3 = (unsigned)(DIMF) << 16;             // dim1 hi=0 | tile_dim0
        g1.s4 = 8u;                                 // tile_dim1 = 8 indices
        g1.s5 = (unsigned)DIMF;                     // tensor_dim0_stride lo
        g1.s6 = 0u;
        g1.s7 = 0u;

        v4i g2, g3;
        g2.x = __builtin_amdgcn_readlane(dval, 0);
        g2.y = __builtin_amdgcn_readlane(dval, 1);
        g2.z = __builtin_amdgcn_readlane(dval, 2);
        g2.w = __builtin_amdgcn_readlane(dval, 3);
        g3.x = __builtin_amdgcn_readlane(dval, 4);
        g3.y = __builtin_amdgcn_readlane(dval, 5);
        g3.z = __builtin_amdgcn_readlane(dval, 6);
        g3.w = __builtin_amdgcn_readlane(dval, 7);
        asm volatile("tensor_load_to_lds %0, %1, %2, %3"
                     :: "s"(g0), "s"(g1), "s"(g2), "s"(g3) : "memory");

        g0.y = ldsbase + 8u * XSTRIDE * 4u;         // rows 8..15
        g2.x = __builtin_amdgcn_readlane(dval, 8);
        g2.y = __builtin_amdgcn_readlane(dval, 9);
        g2.z = __builtin_amdgcn_readlane(dval, 10);
        g2.w = __builtin_amdgcn_readlane(dval, 11);
        g3.x = __builtin_amdgcn_readlane(dval, 12);
        g3.y = __builtin_amdgcn_readlane(dval, 13);
        g3.z = __builtin_amdgcn_readlane(dval, 14);
        g3.w = __builtin_amdgcn_readlane(dval, 15);
        asm volatile("tensor_load_to_lds %0, %1, %2, %3"
                     :: "s"(g0), "s"(g1), "s"(g2), "s"(g3) : "memory");
    }

    // ---- stage per-edge metadata -------------------------------------------
    if (lane < 16) {
        const int e = (int)e0 + lane;
        srcS[wave][lane] = eidx[e];
        const float* p = esh + (long)e * 4;
        shS[wave][lane][0] = p[0];
        shS[wave][lane][1] = p[1];
        shS[wave][lane][2] = p[2];
        shS[wave][lane][3] = p[3];
    }
    __syncthreads();

    // ---- MLP layer 1: h = relu(edge_attr @ fc_w1 + b1)  (16x32 @ 32x128) ---
    // (runs while the TDM gather is in flight)
    {
        v2f aat[8];
        const float* arow = eat + (e0 + r) * EAF;
        #pragma unroll
        for (int s = 0; s < 8; ++s)
            aat[s] = *(const v2f*)(arow + 4 * s + ko);

        #pragma unroll
        for (int t = 0; t < 8; ++t) {                 // 8 N-tiles of 16 cols
            const float bias = fb1[t * 16 + r];
            v8f c;
            #pragma unroll
            for (int i = 0; i < 8; ++i) c[i] = bias;
            #pragma unroll
            for (int s = 0; s < 8; ++s) {             // K = 32, steps of 4
                const int k = 4 * s + ko;
                v2f b;
                b.x = fw1[(k    ) * NEFD + t * 16 + r];
                b.y = fw1[(k + 1) * NEFD + t * 16 + r];
                c = wmma_f32_4(aat[s], b, c);
            }
            #pragma unroll
            for (int i = 0; i < 8; ++i)
                htS[wave][i + 8 * half][t * 16 + r] = fmaxf(c[i], 0.0f);
        }
    }
    __syncthreads();

    // ---- MLP layer 2 (128 -> 4), tiny: plain VALU --------------------------
    for (int o = lane; o < 64; o += 32) {
        const int e = o >> 2, j = o & 3;
        float a = fb2[j];
        #pragma unroll 8
        for (int k = 0; k < NEFD; ++k)
            a = fmaf(htS[wave][e][k], fw2[k * NPATH + j], a);
        wS[wave][e][j] = a;
    }
    __syncthreads();

    // ---- wait for the TDM gather, then read fragments from LDS -------------
    __builtin_amdgcn_s_wait_tensorcnt(0);
    asm volatile("" ::: "memory");
    const float* xt = &xtS[wave][0];

    // x0 A-fragments (16x64)
    v2f ax[16];
    #pragma unroll
    for (int s = 0; s < 16; ++s) {
        ax[s].x = xt[xoff(r, 4 * s + ko)];
        ax[s].y = xt[xoff(r, 4 * s + ko + 1)];
    }

    // ---- C000 = x0 @ W_000  (16x64 @ 64x64) --------------------------------
    v8f c000[4];
    #pragma unroll
    for (int t = 0; t < 4; ++t) {
        v8f c = {0.f, 0.f, 0.f, 0.f, 0.f, 0.f, 0.f, 0.f};
        #pragma unroll
        for (int s = 0; s < 16; ++s) {
            const int k = 4 * s + ko;
            v2f b;
            b.x = W000[(k    ) * MUL0 + t * 16 + r];
            b.y = W000[(k + 1) * MUL0 + t * 16 + r];
            c = wmma_f32_4(ax[s], b, c);
        }
        c000[t] = c;
    }

    // ---- P = x0 @ W_011  (16x64 @ 64x32) -----------------------------------
    v8f P[2];
    #pragma unroll
    for (int t = 0; t < 2; ++t) {
        v8f c = {0.f, 0.f, 0.f, 0.f, 0.f, 0.f, 0.f, 0.f};
        #pragma unroll
        for (int s = 0; s < 16; ++s) {
            const int k = 4 * s + ko;
            v2f b;
            b.x = W011[(k    ) * MUL1 + t * 16 + r];
            b.y = W011[(k + 1) * MUL1 + t * 16 + r];
            c = wmma_f32_4(ax[s], b, c);
        }
        P[t] = c;
    }

    // ---- x1 A-fragments: X1m[r][u] = x[64 + u*3 + m] -----------------------
    v2f a1[3][8];
    #pragma unroll
    for (int s = 0; s < 8; ++s) {
        const int u0 = 4 * s + ko;
        #pragma unroll
        for (int m = 0; m < 3; ++m) {
            v2f v;
            v.x = xt[xoff(r, MUL0 + (u0    ) * 3 + m)];
            v.y = xt[xoff(r, MUL0 + (u0 + 1) * 3 + m)];
            a1[m][s] = v;
        }
    }

    // ---- dot A-fragments: dot[r][u] = (x1 . sh1)/sqrt(3) -------------------
    const float s1x = shS[wave][r][1];
    const float s1y = shS[wave][r][2];
    const float s1z = shS[wave][r][3];
    v2f ad[8];
    #pragma unroll
    for (int s = 0; s < 8; ++s) {
        ad[s].x = INV_SQRT3 * (a1[0][s].x * s1x + a1[1][s].x * s1y + a1[2][s].x * s1z);
        ad[s].y = INV_SQRT3 * (a1[0][s].y * s1x + a1[1][s].y * s1y + a1[2][s].y * s1z);
    }

    // ---- C110 = dot @ W_110  (16x32 @ 32x64) -------------------------------
    v8f c110[4];
    #pragma unroll
    for (int t = 0; t < 4; ++t) {
        v8f c = {0.f, 0.f, 0.f, 0.f, 0.f, 0.f, 0.f, 0.f};
        #pragma unroll
        for (int s = 0; s < 8; ++s) {
            const int k = 4 * s + ko;
            v2f b;
            b.x = W110[(k    ) * MUL0 + t * 16 + r];
            b.y = W110[(k + 1) * MUL0 + t * 16 + r];
            c = wmma_f32_4(ad[s], b, c);
        }
        c110[t] = c;
    }

    // ---- Q[m] = x1_m @ W_101  (16x32 @ 32x32), m = 0..2 --------------------
    v8f Q[3][2];
    #pragma unroll
    for (int m = 0; m < 3; ++m) {
        #pragma unroll
        for (int t = 0; t < 2; ++t) {
            v8f c = {0.f, 0.f, 0.f, 0.f, 0.f, 0.f, 0.f, 0.f};
            #pragma unroll
            for (int s = 0; s < 8; ++s) {
                const int k = 4 * s + ko;
                v2f b;
                b.x = W101[(k    ) * MUL1 + t * 16 + r];
                b.y = W101[(k + 1) * MUL1 + t * 16 + r];
                c = wmma_f32_4(a1[m][s], b, c);
            }
            Q[m][t] = c;
        }
    }

    // ---- combine with per-edge scalars, scatter-add ------------------------
    #pragma unroll
    for (int i = 0; i < 8; ++i) {
        const int me = i + 8 * half;                 // edge row of D-fragment vgpr i
        const float w0 = wS[wave][me][0];
        const float w1 = wS[wave][me][1];
        const float w2 = wS[wave][me][2];
        const float w3 = wS[wave][me][3];
        const float sh0 = shS[wave][me][0];
        const float t1x = shS[wave][me][1];
        const float t1y = shS[wave][me][2];
        const float t1z = shS[wave][me][3];
        const float sA = w0 * sh0 * A000;
        const float sD = w3 * A110;
        const float sB = w1 * A011;
        const float sC = w2 * sh0 * A101;
        float* ob = acc + (long)srcS[wave][me] * DIMF;

        #pragma unroll
        for (int t = 0; t < 4; ++t)                  // out0: features 0..63
            atomicAdd(ob + t * 16 + r, sA * c000[t][i] + sD * c110[t][i]);

        #pragma unroll
        for (int t = 0; t < 2; ++t) {                // out1: features 64 + v*3 + m
            const float pv = sB * P[t][i];
            const int v = t * 16 + r;
            atomicAdd(ob + MUL0 + v * 3 + 0, pv * t1x + sC * Q[0][t][i]);
            atomicAdd(ob + MUL0 + v * 3 + 1, pv * t1y + sC * Q[1][t][i]);
            atomicAdd(ob + MUL0 + v * 3 + 2, pv * t1z + sC * Q[2][t][i]);
        }
    }
    if (lane < 16)
        atomicAdd(cnt + srcS[wave][lane], 1.0f);
}

__global__ __launch_bounds__(256)
void tpconv_norm_kernel(float* __restrict__ out, const float* __restrict__ cnt)
{
    const long i = (long)blockIdx.x * blockDim.x + threadIdx.x;
    if (i < (long)NNODES * DIMF)
        out[i] /= fmaxf(cnt[i / DIMF], 1.0f);
}

extern "C" void kernel_launch(void* const* d_in, const int* in_sizes, int n_in,
                              void* d_out, int out_size, void* d_ws, size_t ws_size,
                              hipStream_t stream) {
    (void)in_sizes; (void)n_in; (void)out_size; (void)ws_size;
    const float* nat  = (const float*)d_in[0];
    const float* eat  = (const float*)d_in[1];
    const float* esh  = (const float*)d_in[2];
    const float* W000 = (const float*)d_in[3];
    const float* W011 = (const float*)d_in[4];
    const float* W101 = (const float*)d_in[5];
    const float* W110 = (const float*)d_in[6];
    const float* fw1  = (const float*)d_in[7];
    const float* fb1  = (const float*)d_in[8];
    const float* fw2  = (const float*)d_in[9];
    const float* fb2  = (const float*)d_in[10];
    const int*   eix  = (const int*)  d_in[11];

    float* out = (float*)d_out;             // accumulate directly into output
    float* cnt = (float*)d_ws;              // [N] counts in workspace

    hipMemsetAsync(out, 0, sizeof(float) * (size_t)NNODES * DIMF, stream);
    hipMemsetAsync(cnt, 0, sizeof(float) * (size_t)NNODES, stream);

    const int edge_blocks = (NEDGES + WAVES * 16 - 1) / (WAVES * 16);   // 25000
    tpconv_edge_kernel<<<edge_blocks, TPB, 0, stream>>>(
        nat, eat, esh, W000, W011, W101, W110, fw1, fb1, fw2, fb2, eix, out, cnt);

    const long n = (long)NNODES * DIMF;
    tpconv_norm_kernel<<<(int)((n + 255) / 256), 256, 0, stream>>>(out, cnt);
}